// GCN_85255100825815
// MI455X (gfx1250) — compile-verified
//
#include <hip/hip_runtime.h>
#include <hip/hip_bf16.h>

#define IN_DIM 256
#define OUT_DIM 256
#define BN_EPS 1e-5f

typedef __attribute__((ext_vector_type(2))) float v2f;
typedef __attribute__((ext_vector_type(8))) float v8f;

__device__ __forceinline__ void atomic_add_f32(float* p, float v) {
    // relaxed, device scope -> hardware global_atomic_add_f32 (no CAS loop)
    __hip_atomic_fetch_add(p, v, __ATOMIC_RELAXED, __HIP_MEMORY_SCOPE_AGENT);
}

// ---------------------------------------------------------------------------
// zero-fill (float4 granularity, grid-stride)
// ---------------------------------------------------------------------------
__global__ __launch_bounds__(256) void zero_f4(float4* __restrict__ p, long n4) {
    long i = (long)blockIdx.x * blockDim.x + threadIdx.x;
    long stride = (long)gridDim.x * blockDim.x;
    float4 zv = make_float4(0.f, 0.f, 0.f, 0.f);
    for (; i < n4; i += stride) p[i] = zv;
}

// ---------------------------------------------------------------------------
// z = x @ W  via V_WMMA_F32_16X16X4_F32.
// One wave computes a 16x64 slab: 4 accumulators sharing one A fragment per
// K-step (4x A-reuse). Block = 8 waves = 2 row tiles x all 256 columns.
// grid.x = ceil(n/32).
// A(16x4 f32): lane l(0..15): M=l, VGPR0=K0, VGPR1=K1; lanes 16..31: K2,K3.
// B(4x16 f32): lane l(0..15): N=l, VGPR0=K0, VGPR1=K1; lanes 16..31: K2,K3.
// C/D: VGPR r, lanes 0..15 -> (M=r, N=lane); lanes 16..31 -> (M=8+r, N=l).
// ---------------------------------------------------------------------------
__global__ __launch_bounds__(256) void gemm_wmma_f32(const float* __restrict__ x,
                                                     const float* __restrict__ w,
                                                     float* __restrict__ z, int n) {
    const int lane = threadIdx.x & 31;
    const int wv   = threadIdx.x >> 5;       // 0..7
    const int wrow = wv >> 2;                // 0..1 : row tile within block
    const int wcol = wv & 3;                 // 0..3 : 64-column slab
    const int row0 = (blockIdx.x * 2 + wrow) * 16;
    const int col0 = wcol * 64;
    const int half = lane >> 4;              // 0 or 1
    const int l    = lane & 15;

    if (row0 >= n) return;                   // uniform per wave
    int rr = row0 + l;
    if (rr >= n) rr = n - 1;                 // clamp tail-tile row reads
    const float* __restrict__ xrow = x + (size_t)rr * IN_DIM;

    v8f acc0 = {}, acc1 = {}, acc2 = {}, acc3 = {};

#if __has_builtin(__builtin_amdgcn_wmma_f32_16x16x4_f32)
#pragma unroll 4
    for (int k = 0; k < IN_DIM; k += 4) {
        const int ka = k + 2 * half;         // this lane's K pair
        v2f a;
        a.x = xrow[ka];
        a.y = xrow[ka + 1];

        const float* __restrict__ wr0 = w + (size_t)ka * OUT_DIM + col0 + l;
        const float* __restrict__ wr1 = wr0 + OUT_DIM;
        v2f b0, b1, b2, b3;
        b0.x = wr0[0];  b0.y = wr1[0];
        b1.x = wr0[16]; b1.y = wr1[16];
        b2.x = wr0[32]; b2.y = wr1[32];
        b3.x = wr0[48]; b3.y = wr1[48];

        acc0 = __builtin_amdgcn_wmma_f32_16x16x4_f32(false, a, false, b0, (short)0, acc0, false, false);
        acc1 = __builtin_amdgcn_wmma_f32_16x16x4_f32(false, a, false, b1, (short)0, acc1, false, false);
        acc2 = __builtin_amdgcn_wmma_f32_16x16x4_f32(false, a, false, b2, (short)0, acc2, false, false);
        acc3 = __builtin_amdgcn_wmma_f32_16x16x4_f32(false, a, false, b3, (short)0, acc3, false, false);
    }
#else
    // scalar fallback (kept only so a missing builtin cannot hard-fail the build)
    for (int k = 0; k < IN_DIM; ++k) {
        float av = xrow[k];
        for (int r = 0; r < 8; ++r) {
            acc0[r] += av * w[(size_t)k * OUT_DIM + col0 + l];
            acc1[r] += av * w[(size_t)k * OUT_DIM + col0 + 16 + l];
            acc2[r] += av * w[(size_t)k * OUT_DIM + col0 + 32 + l];
            acc3[r] += av * w[(size_t)k * OUT_DIM + col0 + 48 + l];
        }
    }
#endif

    float* __restrict__ zt = z + (size_t)row0 * OUT_DIM + col0 + l;
    if (row0 + 16 <= n) {                    // uniform: full tile, no lane guards
#pragma unroll
        for (int r = 0; r < 8; ++r) {
            const int m = r + 8 * half;
            float* p = zt + (size_t)m * OUT_DIM;
            p[0]  = acc0[r];
            p[16] = acc1[r];
            p[32] = acc2[r];
            p[48] = acc3[r];
        }
    } else {                                 // tail tile (unused when n % 16 == 0)
#pragma unroll
        for (int r = 0; r < 8; ++r) {
            const int m = r + 8 * half;
            if (row0 + m < n) {
                float* p = zt + (size_t)m * OUT_DIM;
                p[0]  = acc0[r];
                p[16] = acc1[r];
                p[32] = acc2[r];
                p[48] = acc3[r];
            }
        }
    }
}

// ---------------------------------------------------------------------------
// SpMM scatter: agg[row] += val * z[col]. 4 edges per 256-thread block,
// 64 lanes per edge, float4 per lane (coalesced b128 gather, L2-resident z).
// ---------------------------------------------------------------------------
__global__ __launch_bounds__(256) void spmm_scatter(const int*   __restrict__ erows,
                                                    const int*   __restrict__ ecols,
                                                    const float* __restrict__ evals,
                                                    const float* __restrict__ z,
                                                    float* __restrict__ agg, int E) {
    int e = blockIdx.x * 4 + (threadIdx.x >> 6);
    if (e >= E) return;
    const int t = threadIdx.x & 63;

    const int   r = erows[e];
    const int   c = ecols[e];
    const float v = evals[e];

    const float4 m = ((const float4*)(z + (size_t)c * OUT_DIM))[t];
    float* o = agg + (size_t)r * OUT_DIM + t * 4;
    atomic_add_f32(o + 0, v * m.x);
    atomic_add_f32(o + 1, v * m.y);
    atomic_add_f32(o + 2, v * m.z);
    atomic_add_f32(o + 3, v * m.w);
}

// ---------------------------------------------------------------------------
// Column-wise sum / sum-of-squares. Thread t owns column t; rows are strided
// across blocks so reads stay coalesced. Partial results atomically reduced.
// ---------------------------------------------------------------------------
__global__ __launch_bounds__(256) void bn_stats(const float* __restrict__ agg,
                                                float* __restrict__ sums,
                                                float* __restrict__ sqs, int n) {
    const int c = threadIdx.x;               // 0..255 == column
    float s = 0.f, q = 0.f;
    for (int r = blockIdx.x; r < n; r += gridDim.x) {
        float v = agg[(size_t)r * OUT_DIM + c];
        s += v;
        q += v * v;
    }
    atomic_add_f32(&sums[c], s);
    atomic_add_f32(&sqs[c], q);
}

// ---------------------------------------------------------------------------
// Fused BatchNorm (biased var) + ReLU, in place over agg -> out.
// ---------------------------------------------------------------------------
__global__ __launch_bounds__(256) void bn_relu(const float* __restrict__ agg,
                                               const float* __restrict__ sums,
                                               const float* __restrict__ sqs,
                                               const float* __restrict__ gamma,
                                               const float* __restrict__ beta,
                                               float* __restrict__ out, int n) {
    const float invN = 1.0f / (float)n;
    long i = (long)blockIdx.x * blockDim.x + threadIdx.x;
    long total = (long)n * OUT_DIM;
    long stride = (long)gridDim.x * blockDim.x;
    for (; i < total; i += stride) {
        int c = (int)(i & (OUT_DIM - 1));
        float mean = sums[c] * invN;
        float var  = sqs[c] * invN - mean * mean;
        float rs   = rsqrtf(var + BN_EPS);
        float v    = (agg[i] - mean) * rs * gamma[c] + beta[c];
        out[i] = v > 0.f ? v : 0.f;
    }
}

// ---------------------------------------------------------------------------
extern "C" void kernel_launch(void* const* d_in, const int* in_sizes, int n_in,
                              void* d_out, int out_size, void* d_ws, size_t ws_size,
                              hipStream_t stream) {
    const float* x     = (const float*)d_in[0];
    const int*   erows = (const int*)  d_in[1];
    const int*   ecols = (const int*)  d_in[2];
    const float* evals = (const float*)d_in[3];
    const float* w     = (const float*)d_in[4];
    const float* gamma = (const float*)d_in[5];
    const float* beta  = (const float*)d_in[6];
    float*       out   = (float*)d_out;

    const int n = in_sizes[0] / IN_DIM;   // 100000
    const int E = in_sizes[1];            // 3200000

    // workspace: z [n*256] f32, then sums[256], sqs[256]
    float* z    = (float*)d_ws;
    float* sums = z + (size_t)n * OUT_DIM;
    float* sqs  = sums + OUT_DIM;

    // agg accumulates directly into d_out (zeroed each call; harness poisons it)
    zero_f4<<<4096, 256, 0, stream>>>((float4*)out, (long)n * OUT_DIM / 4);
    zero_f4<<<1, 256, 0, stream>>>((float4*)sums, (2 * OUT_DIM) / 4);

    gemm_wmma_f32<<<(n + 31) / 32, 256, 0, stream>>>(x, w, z, n);

    spmm_scatter<<<(E + 3) / 4, 256, 0, stream>>>(erows, ecols, evals, z, out, E);

    bn_stats<<<512, 256, 0, stream>>>(out, sums, sqs, n);
    bn_relu<<<4096, 256, 0, stream>>>(out, sums, sqs, gamma, beta, out, n);
}